// MessagePassingConvolution_80401787781509
// MI455X (gfx1250) — compile-verified
//
#include <hip/hip_runtime.h>

typedef __attribute__((ext_vector_type(2))) float v2f;
typedef __attribute__((ext_vector_type(8))) float v8f;

#define WAVES_PER_BLOCK 8
#define EDGES_PER_WAVE  16
#define EDGES_PER_BLOCK (WAVES_PER_BLOCK * EDGES_PER_WAVE)

__device__ __forceinline__ void atomic_add_f32(float* p, float v) {
  __hip_atomic_fetch_add(p, v, __ATOMIC_RELAXED, __HIP_MEMORY_SCOPE_AGENT);
}

__global__ void __launch_bounds__(256) zero_kernel(float* __restrict__ out, int n) {
  int i = blockIdx.x * 256 + threadIdx.x;
  if (i < n) out[i] = 0.0f;
}

// One wave handles a 16-edge tile:
//   GEMM1: h = radial(16x8) @ w1(8x8, padded to 8x16)   -> 2x v_wmma_f32_16x16x4_f32
//   swish (x * v_rcp_f32(1+exp(-x))), LDS re-layout (C-layout -> A-layout)
//   GEMM2: w = h(16x8) @ w2(8x48)                        -> 6x v_wmma_f32_16x16x4_f32
//   scatter: lane = (edge m = lane&15, half = lane>>4), 48 fp32 atomics per lane
__global__ void __launch_bounds__(256)
mp_conv_kernel(const float* __restrict__ node_feats,   // (N,32)
               const float* __restrict__ edge_feats,   // (E,4)
               const float* __restrict__ radial,       // (E,8)
               const float* __restrict__ w1,           // (8,8)
               const float* __restrict__ w2,           // (8,48)
               const int*   __restrict__ senders,
               const int*   __restrict__ receivers,
               float* __restrict__ out,                // (N,96)
               int E) {
  __shared__ float lds_h[WAVES_PER_BLOCK][16 * 8];
  __shared__ float lds_w[WAVES_PER_BLOCK][16 * 48];

  const int lane = threadIdx.x & 31;
  const int wave = threadIdx.x >> 5;
  const int n    = lane & 15;   // WMMA row (edge-in-tile) / column index
  const int hh   = lane >> 4;   // lane half: selects K pair in A/B layouts
  const int tileBase = (blockIdx.x * WAVES_PER_BLOCK + wave) * EDGES_PER_WAVE;
  const int e  = tileBase + n;
  const int ec = (e < E) ? e : (E - 1);

  // ---------- GEMM1: A = radial tile, 16x4 f32 A-layout:
  // lane L holds row M=L&15; VGPR0: k = 2*hh, VGPR1: k = 2*hh+1 (chunk adds +4)
  const float4 r0 = *(const float4*)(radial + (size_t)ec * 8);
  const float4 r1 = *(const float4*)(radial + (size_t)ec * 8 + 4);
  v2f a0, a1;
  a0.x = hh ? r0.z : r0.x;  a0.y = hh ? r0.w : r0.y;   // k = 2hh, 2hh+1
  a1.x = hh ? r1.z : r1.x;  a1.y = hh ? r1.w : r1.y;   // k = 4+2hh, 5+2hh
  // B (4x16): col n = lane&15; VGPR v holds k = chunk*4 + 2*hh + v. Pad cols >= 8 with 0.
  const int   nk = (n < 8) ? n : 0;
  const float bz = (n < 8) ? 1.0f : 0.0f;
  v2f b0, b1;
  b0.x = bz * w1[(2 * hh + 0) * 8 + nk];
  b0.y = bz * w1[(2 * hh + 1) * 8 + nk];
  b1.x = bz * w1[(4 + 2 * hh + 0) * 8 + nk];
  b1.y = bz * w1[(4 + 2 * hh + 1) * 8 + nk];

  v8f h = {};
  h = __builtin_amdgcn_wmma_f32_16x16x4_f32(false, a0, false, b0, (short)0, h, false, false);
  h = __builtin_amdgcn_wmma_f32_16x16x4_f32(false, a1, false, b1, (short)0, h, false, false);

  // swish via fast reciprocal: denominator >= 1, v_rcp_f32 (~1 ulp) is plenty.
  // x -> -inf: exp(-x)=inf, rcp=0, x*0=0 (correct limit). Cols >= 8 are exactly 0, stay 0.
  #pragma unroll
  for (int g = 0; g < 8; ++g) {
    float x = h[g];
    h[g] = x * __builtin_amdgcn_rcpf(1.0f + __expf(-x));
  }

  // C-layout -> row-major 16x8 in per-wave LDS (element (g+8*hh, n) for n<8)
  float* hbuf = lds_h[wave];
  if (n < 8) {
    #pragma unroll
    for (int g = 0; g < 8; ++g)
      hbuf[(g + 8 * hh) * 8 + n] = h[g];
  }
  asm volatile("s_wait_dscnt 0" ::: "memory");

  // ---------- GEMM2: A = h (16x8) in A-layout from LDS
  v2f a20, a21;
  a20.x = hbuf[n * 8 + 2 * hh + 0];
  a20.y = hbuf[n * 8 + 2 * hh + 1];
  a21.x = hbuf[n * 8 + 4 + 2 * hh + 0];
  a21.y = hbuf[n * 8 + 4 + 2 * hh + 1];

  float* wbuf = lds_w[wave];
  #pragma unroll
  for (int t = 0; t < 3; ++t) {            // 48 = 3 x 16 output columns
    v2f b20, b21;
    b20.x = w2[(2 * hh + 0) * 48 + t * 16 + n];
    b20.y = w2[(2 * hh + 1) * 48 + t * 16 + n];
    b21.x = w2[(4 + 2 * hh + 0) * 48 + t * 16 + n];
    b21.y = w2[(4 + 2 * hh + 1) * 48 + t * 16 + n];
    v8f acc = {};
    acc = __builtin_amdgcn_wmma_f32_16x16x4_f32(false, a20, false, b20, (short)0, acc, false, false);
    acc = __builtin_amdgcn_wmma_f32_16x16x4_f32(false, a21, false, b21, (short)0, acc, false, false);
    #pragma unroll
    for (int g = 0; g < 8; ++g)
      wbuf[(g + 8 * hh) * 48 + t * 16 + n] = acc[g];
  }
  asm volatile("s_wait_dscnt 0" ::: "memory");

  // ---------- tensor product + scatter (after all WMMAs; divergence OK now)
  if (e >= E) return;

  const int snd = senders[e];
  const int rcv = receivers[e];
  const float4* nf = (const float4*)(node_feats + (size_t)snd * 32);
  float4 q0 = nf[0], q1 = nf[1];
  float s[8] = {q0.x, q0.y, q0.z, q0.w, q1.x, q1.y, q1.z, q1.w};
  float v[24];
  #pragma unroll
  for (int i = 0; i < 6; ++i) {
    float4 q = nf[2 + i];
    v[i*4+0] = q.x; v[i*4+1] = q.y; v[i*4+2] = q.z; v[i*4+3] = q.w;
  }
  const float4 ef = *(const float4*)(edge_feats + (size_t)e * 4);
  const float e0 = ef.x, e1x = ef.y, e1y = ef.z, e1z = ef.w;
  const float* w = wbuf + n * 48;
  float* op = out + (size_t)rcv * 96 + hh * 48;
  const float SC  = 0.17677669529663687f;   // 1/sqrt(32)
  const float IS3 = 0.57735026918962576f;   // 1/sqrt(3)

  if (hh == 0) {
    // components 0..47: scal(24) + vec rows 0..7 (v * w[24+r])
    #pragma unroll
    for (int j = 0; j < 8; ++j) atomic_add_f32(op + j, s[j] * w[j] * SC);
    #pragma unroll
    for (int j = 0; j < 8; ++j) atomic_add_f32(op + 8 + j, s[j] * e0 * w[8 + j] * SC);
    #pragma unroll
    for (int j = 0; j < 8; ++j) {
      float d = (v[3*j] * e1x + v[3*j+1] * e1y + v[3*j+2] * e1z) * IS3;
      atomic_add_f32(op + 16 + j, d * w[16 + j] * SC);
    }
    #pragma unroll
    for (int r = 0; r < 8; ++r) {
      float wr = w[24 + r] * SC;
      atomic_add_f32(op + 24 + r*3 + 0, v[r*3+0] * wr);
      atomic_add_f32(op + 24 + r*3 + 1, v[r*3+1] * wr);
      atomic_add_f32(op + 24 + r*3 + 2, v[r*3+2] * wr);
    }
  } else {
    // components 48..95: vec rows 8..15 (s x e1 * w[32+r]) and 16..23 (v * e0 * w[40+r])
    #pragma unroll
    for (int r = 0; r < 8; ++r) {
      float wr = w[32 + r] * SC;
      atomic_add_f32(op + r*3 + 0, s[r] * e1x * wr);
      atomic_add_f32(op + r*3 + 1, s[r] * e1y * wr);
      atomic_add_f32(op + r*3 + 2, s[r] * e1z * wr);
    }
    #pragma unroll
    for (int r = 0; r < 8; ++r) {
      float wr = w[40 + r] * e0 * SC;
      atomic_add_f32(op + 24 + r*3 + 0, v[r*3+0] * wr);
      atomic_add_f32(op + 24 + r*3 + 1, v[r*3+1] * wr);
      atomic_add_f32(op + 24 + r*3 + 2, v[r*3+2] * wr);
    }
  }
}

extern "C" void kernel_launch(void* const* d_in, const int* in_sizes, int n_in,
                              void* d_out, int out_size, void* d_ws, size_t ws_size,
                              hipStream_t stream) {
  const float* node_feats = (const float*)d_in[0];
  const float* edge_feats = (const float*)d_in[1];
  const float* radial     = (const float*)d_in[2];
  const float* w1         = (const float*)d_in[3];
  const float* w2         = (const float*)d_in[4];
  const int*   senders    = (const int*)d_in[5];
  const int*   receivers  = (const int*)d_in[6];
  float* out = (float*)d_out;

  // d_out is poisoned by the harness; segment_sum accumulates from zero.
  zero_kernel<<<(out_size + 255) / 256, 256, 0, stream>>>(out, out_size);

  const int E = in_sizes[5];                       // senders element count
  const int blocks = (E + EDGES_PER_BLOCK - 1) / EDGES_PER_BLOCK;
  mp_conv_kernel<<<blocks, 256, 0, stream>>>(node_feats, edge_feats, radial, w1, w2,
                                             senders, receivers, out, E);
}